// ZINBDecoder_32607391711809
// MI455X (gfx1250) — compile-verified
//
#include <hip/hip_runtime.h>
#include <hip/hip_bf16.h>
#include <math.h>

typedef float v2f __attribute__((ext_vector_type(2)));
typedef float v8f __attribute__((ext_vector_type(8)));

#define WAVES_PER_BLOCK 8
#define BLOCK_THREADS (WAVES_PER_BLOCK * 32)

// ZINB decoder over a cell-gene edge list.
// Per 16-edge tile (one wave32): H_tile[16x64] = c_feat[u]*g_feat[v] (gathered,
// laid out directly in WMMA A format), dots = H_tile @ W[64x3] via 16 chained
// V_WMMA_F32_16X16X4_F32, epilogue on lanes 0-15.
__global__ __launch_bounds__(BLOCK_THREADS)
void zinb_decoder_wmma(const float* __restrict__ c_feat,
                       const float* __restrict__ g_feat,
                       const float* __restrict__ cs_factor,
                       const float* __restrict__ gs_factor,
                       const int*   __restrict__ edge_u,
                       const int*   __restrict__ edge_v,
                       const float* __restrict__ W_mean, const float* __restrict__ b_mean,
                       const float* __restrict__ W_disp, const float* __restrict__ b_disp,
                       const float* __restrict__ W_pi,   const float* __restrict__ b_pi,
                       float* __restrict__ out, int E)
{
    __shared__ float lds_dots[WAVES_PER_BLOCK * 48];   // per-wave 3x16 dot staging

    const int lane  = threadIdx.x & 31;
    const int wave  = threadIdx.x >> 5;
    const long gwave  = (long)blockIdx.x * WAVES_PER_BLOCK + wave;
    const long nwaves = (long)gridDim.x * WAVES_PER_BLOCK;

    const int halfSel = (lane < 16) ? 0 : 2;  // K offset inside each K=4 chunk
    const int col     = lane & 15;            // N column of B / edge id in tile

    // ---- Build B (4x16, K x N) registers, held for the whole kernel ----
    // col 0 = W_mean, col 1 = W_disp, col 2 = W_pi, cols 3..15 = 0.
    // Layout: VGPR0 = {K=0 lanes0-15, K=2 lanes16-31}; VGPR1 = {K=1, K=3}.
    v2f Bk[16];
    {
        const float* Wsel = (col == 0) ? W_mean : (col == 1) ? W_disp : W_pi;
#pragma unroll
        for (int k = 0; k < 16; ++k) {
            float w0 = 0.f, w1 = 0.f;
            if (col < 3) {
                w0 = Wsel[4 * k + halfSel];
                w1 = Wsel[4 * k + halfSel + 1];
            }
            v2f b; b.x = w0; b.y = w1;
            Bk[k] = b;
        }
    }
    const float bm = b_mean[0];
    const float bd = b_disp[0];
    const float bp = b_pi[0];

    const long ntiles = ((long)E + 15) >> 4;
    for (long tile = gwave; tile < ntiles; tile += nwaves) {   // wave-uniform loop
        const long base = tile << 4;
        const int  eidx   = (int)base + col;
        const bool valid  = (eidx < E);
        const int  eclamp = valid ? eidx : (E - 1);

        const int u = edge_u[eclamp];
        const int v = edge_v[eclamp];
        const float* crow = c_feat + (long)u * 64 + halfSel;
        const float* grow = g_feat + (long)v * 64 + halfSel;

        v8f acc = {0.f, 0.f, 0.f, 0.f, 0.f, 0.f, 0.f, 0.f};
#pragma unroll
        for (int k = 0; k < 16; ++k) {
            // float2 gather lands exactly in 32-bit A-matrix 16x4 layout
            v2f cv = *(const v2f*)(crow + 4 * k);
            v2f gv = *(const v2f*)(grow + 4 * k);
            v2f a; a.x = cv.x * gv.x; a.y = cv.y * gv.y;
            acc = __builtin_amdgcn_wmma_f32_16x16x4_f32(
                false, a, false, Bk[k], (short)0, acc, false, false);
        }

        // Stage useful D columns (N=0..2) to LDS.
        // D[j] on lane l = element (M = j + (l<16 ? 0 : 8), N = l&15).
        float* myLds = lds_dots + wave * 48;
        if (col < 3) {
            const int rbase = (lane < 16) ? 0 : 8;
#pragma unroll
            for (int j = 0; j < 8; ++j)
                myLds[col * 16 + rbase + j] = acc[j];
        }
        asm volatile("s_wait_dscnt 0x0" ::: "memory");

        if ((lane < 16) & valid) {
            const float dm = myLds[lane];        // dot with W_mean for edge `lane`
            const float dd = myLds[16 + lane];   // dot with W_disp
            const float dp = myLds[32 + lane];   // dot with W_pi
            const float cs = cs_factor[u];       // u/v already = this edge's nodes
            const float gs = gs_factor[v];

            const float mu_s = 1.f / (1.f + expf(-(dm + bm)));       // sigmoid
            const float pi   = 1.f / (1.f + expf(-(dp + bp)));       // sigmoid
            const float x    = gs * (dd + bd);
            const float sp   = fmaxf(x, 0.f) + log1pf(expf(-fabsf(x))); // softplus
            const float disp = fminf(fmaxf(sp, 1e-4f), 1e4f);
            const float mu   = cs * fminf(fmaxf(expf(gs * mu_s) - 1.f, 1e-5f), 1e6f);

            out[eidx]             = mu;
            out[(long)E + eidx]   = disp;
            out[2L * (long)E + eidx] = pi;
        }
    }
}

extern "C" void kernel_launch(void* const* d_in, const int* in_sizes, int n_in,
                              void* d_out, int out_size, void* d_ws, size_t ws_size,
                              hipStream_t stream) {
    const float* c_feat    = (const float*)d_in[0];
    const float* g_feat    = (const float*)d_in[1];
    const float* cs_factor = (const float*)d_in[2];
    const float* gs_factor = (const float*)d_in[3];
    const int*   edge_u    = (const int*)d_in[4];
    const int*   edge_v    = (const int*)d_in[5];
    const float* W_mean    = (const float*)d_in[6];
    const float* b_mean    = (const float*)d_in[7];
    const float* W_disp    = (const float*)d_in[8];
    const float* b_disp    = (const float*)d_in[9];
    const float* W_pi      = (const float*)d_in[10];
    const float* b_pi      = (const float*)d_in[11];
    float* out = (float*)d_out;

    const int E = in_sizes[4];
    // 1024 blocks * 8 waves = 8192 waves; ~23 16-edge tiles per wave at E=3M.
    const int grid = 1024;
    zinb_decoder_wmma<<<grid, BLOCK_THREADS, 0, stream>>>(
        c_feat, g_feat, cs_factor, gs_factor, edge_u, edge_v,
        W_mean, b_mean, W_disp, b_disp, W_pi, b_pi, out, E);
}